// ST_CPAB_54468775248450
// MI455X (gfx1250) — compile-verified
//
#include <hip/hip_runtime.h>
#include <stdint.h>

// CPAB spatial transformer, fused: A-matrix build + 50-step Euler integration
// + bilinear gather via CDNA5 async global->LDS, 3-deep double buffering.
// B=8, H=W=256, C=64, NX=2, NY=4, NT=32, NSTEPS=50.

namespace {
constexpr int Hc = 256;
constexpr int Wc = 256;
constexpr int Cc = 64;
constexpr int NSTEPSc = 50;
constexpr int ROWBYTES = Wc * Cc * 4;   // 65536: byte stride between image rows
}

__global__ __launch_bounds__(256) void st_cpab_kernel(
    const float* __restrict__ xin,
    const float* __restrict__ theta,
    float* __restrict__ out)
{
    __shared__ float    sA[32 * 6];        // 32 triangles x (2x3) affine matrix
    __shared__ uint32_t sBase[256];        // per-pixel byte offset of corner (y0,x0) row
    __shared__ float    sWx[256];
    __shared__ float    sWy[256];
    __shared__ __align__(16) float sBuf[8 * 4 * 256];  // 8 waves x 4 buffers x 1KB

    const int tid  = threadIdx.x;
    const int b    = blockIdx.x >> 8;         // batch
    const int pix0 = (blockIdx.x & 255) << 8; // first pixel of this block

    // ---------------- Stage A matrices into LDS (threads 0..31) ----------------
    if (tid < 32) {
        const int t = tid;
        const int i = (t >> 2) & 1;
        const int j = t >> 3;
        const int k = t & 3;
        // quad corner vertex ids, cyclic order ll -> lr -> ur -> ul
        const int ord[4] = { j * 3 + i, j * 3 + i + 1, (j + 1) * 3 + i + 1, (j + 1) * 3 + i };
        const int vid[3] = { ord[k], ord[(k + 1) & 3], 15 + j * 2 + i };

        double px[3], py[3];
        float  velx[3], vely[3];
        for (int n = 0; n < 3; ++n) {
            const int idx = vid[n];
            if (idx < 15) { px[n] = (idx % 3) * 0.5;  py[n] = (idx / 3) * 0.25; }
            else { const int q = idx - 15; px[n] = (q & 1) * 0.5 + 0.25; py[n] = (q >> 1) * 0.25 + 0.125; }
            const int slot = (idx == 4) ? 0 : (idx == 7) ? 1 : (idx == 10) ? 2
                           : (idx >= 15) ? (idx - 12) : -1;
            velx[n] = (slot >= 0) ? theta[b * 22 + 2 * slot]     : 0.0f;
            vely[n] = (slot >= 0) ? theta[b * 22 + 2 * slot + 1] : 0.0f;
        }
        // invP of [[x0,x1,x2],[y0,y1,y2],[1,1,1]] in f64, rounded to f32 (as reference)
        const double X0 = px[0], X1 = px[1], X2 = px[2];
        const double Y0 = py[0], Y1 = py[1], Y2 = py[2];
        const double det = X0 * (Y1 - Y2) - X1 * (Y0 - Y2) + X2 * (Y0 - Y1);
        const double r = 1.0 / det;
        float inv[3][3];
        inv[0][0] = (float)((Y1 - Y2) * r); inv[0][1] = (float)((X2 - X1) * r); inv[0][2] = (float)((X1 * Y2 - X2 * Y1) * r);
        inv[1][0] = (float)((Y2 - Y0) * r); inv[1][1] = (float)((X0 - X2) * r); inv[1][2] = (float)((X2 * Y0 - X0 * Y2) * r);
        inv[2][0] = (float)((Y0 - Y1) * r); inv[2][1] = (float)((X1 - X0) * r); inv[2][2] = (float)((X0 * Y1 - X1 * Y0) * r);
        for (int w = 0; w < 3; ++w) {
            sA[t * 6 + w]     = velx[0] * inv[0][w] + velx[1] * inv[1][w] + velx[2] * inv[2][w];
            sA[t * 6 + 3 + w] = vely[0] * inv[0][w] + vely[1] * inv[1][w] + vely[2] * inv[2][w];
        }
    }
    __syncthreads();

    // ---------------- Phase 1: integrate one pixel per thread ----------------
    {
        const int n  = pix0 + tid;
        const int hh = n >> 8;
        const int ww = n & 255;
        float pxf = (float)ww * (1.0f / 255.0f);
        float pyf = (float)hh * (1.0f / 255.0f);
        const float dt = 1.0f / (float)NSTEPSc;

        for (int s = 0; s < NSTEPSc; ++s) {
            const float xx = fminf(fmaxf(pxf, 0.0f), 1.0f) * 2.0f;
            const float yy = fminf(fmaxf(pyf, 0.0f), 1.0f) * 4.0f;
            const float cx = fminf(fmaxf(floorf(xx), 0.0f), 1.0f);
            const float cy = fminf(fmaxf(floorf(yy), 0.0f), 3.0f);
            const float xr = xx - cx;
            const float yr = yy - cy;
            const bool bd = (yr <= xr);
            const bool ba = (yr <= 1.0f - xr);
            const int tri  = bd ? (ba ? 0 : 1) : (ba ? 3 : 2);
            const int cell = 4 * ((int)cx + ((int)cy << 1)) + tri;
            const float* Ai = &sA[cell * 6];
            const float vx = Ai[0] * pxf + Ai[1] * pyf + Ai[2];
            const float vy = Ai[3] * pxf + Ai[4] * pyf + Ai[5];
            pxf += dt * vx;
            pyf += dt * vy;
        }
        // bilinear params
        const float gx  = fminf(fmaxf(pxf, 0.0f), 1.0f) * 255.0f;
        const float gy  = fminf(fmaxf(pyf, 0.0f), 1.0f) * 255.0f;
        const float fx0 = fminf(floorf(gx), 254.0f);
        const float fy0 = fminf(floorf(gy), 254.0f);
        const int ix0 = (int)fx0;
        const int iy0 = (int)fy0;
        sWx[tid]   = gx - fx0;
        sWy[tid]   = gy - fy0;
        sBase[tid] = (uint32_t)((((b * Hc + iy0) * Wc + ix0) * Cc) * 4);
    }

    // ---------------- Phase 2: per-wave async-gather pipeline + blend ----------------
    // Each wave consumes exactly the 32 pixels its own lanes produced above
    // (same-wave LDS producer/consumer: ordered by DScnt, no barrier needed).
    const int wv   = tid >> 5;
    const int lane = tid & 31;
    const uint32_t laneByte = (uint32_t)lane * 16u;                       // 32 lanes x b128 = 512B
    const uint64_t xbase    = (uint64_t)(uintptr_t)xin;
    const uint32_t ldsBuf0  = (uint32_t)(uintptr_t)(&sBuf[wv * 4 * 256]); // LDS byte offset of buffer 0

    // Issue pixel p's two 512B spans (row y0, row y0+1) into buffer (p&3).
    auto issue = [&](int p) {
        const uint32_t base = sBase[wv * 32 + p];
        const uint64_t ga0 = xbase + base + laneByte;
        const uint64_t ga1 = ga0 + (uint64_t)ROWBYTES;
        const uint32_t la0 = ldsBuf0 + (uint32_t)((p & 3) * 1024) + laneByte;
        const uint32_t la1 = la0 + 512u;
        asm volatile("global_load_async_to_lds_b128 %0, %1, off" :: "v"(la0), "v"(ga0) : "memory");
        asm volatile("global_load_async_to_lds_b128 %0, %1, off" :: "v"(la1), "v"(ga1) : "memory");
    };

    // Blend pixel p from buffer (p&3); each lane handles channels 2*lane, 2*lane+1.
    auto blend = [&](int p) {
        const float* bufp = &sBuf[(wv * 4 + (p & 3)) * 256];
        const float wxp = sWx[wv * 32 + p];
        const float wyp = sWy[wv * 32 + p];
        const int   c0  = lane * 2;

        const float2 v00 = *(const float2*)(bufp + c0);        // (y0  , x0  )
        const float2 v01 = *(const float2*)(bufp + 64 + c0);   // (y0  , x0+1)
        const float2 v10 = *(const float2*)(bufp + 128 + c0);  // (y0+1, x0  )
        const float2 v11 = *(const float2*)(bufp + 192 + c0);  // (y0+1, x0+1)

        const float w00 = (1.0f - wxp) * (1.0f - wyp);
        const float w01 = wxp * (1.0f - wyp);
        const float w10 = (1.0f - wxp) * wyp;
        const float w11 = wxp * wyp;

        float2 o;
        o.x = v00.x * w00 + v01.x * w01 + v10.x * w10 + v11.x * w11;
        o.y = v00.y * w00 + v01.y * w01 + v10.y * w10 + v11.y * w11;

        const size_t npix = (size_t)b * (Hc * Wc) + (size_t)(pix0 + wv * 32 + p);
        *(float2*)(out + npix * Cc + c0) = o;
    };

    // Prologue: 3 pixels in flight (6 async ops outstanding).
    issue(0);
    issue(1);
    issue(2);

    // Steady state: issue p+3, wait until pixel p's 2 loads have landed
    // (in-order completion: asynccnt <= 6 with 8 outstanding => oldest pixel done).
    for (int p = 0; p < 29; ++p) {
        // buffer (p+3)&3 was last read at iteration p-1; its ds reads are done.
        asm volatile("s_wait_dscnt 0x0" ::: "memory");
        issue(p + 3);
        asm volatile("s_wait_asynccnt 0x6" ::: "memory");
        blend(p);
    }
    // Epilogue drain.
    asm volatile("s_wait_asynccnt 0x4" ::: "memory");
    blend(29);
    asm volatile("s_wait_asynccnt 0x2" ::: "memory");
    blend(30);
    asm volatile("s_wait_asynccnt 0x0" ::: "memory");
    blend(31);
}

extern "C" void kernel_launch(void* const* d_in, const int* in_sizes, int n_in,
                              void* d_out, int out_size, void* d_ws, size_t ws_size,
                              hipStream_t stream) {
    (void)in_sizes; (void)n_in; (void)out_size; (void)d_ws; (void)ws_size;
    const float* x     = (const float*)d_in[0];
    const float* theta = (const float*)d_in[1];
    float* out = (float*)d_out;
    // 8 batches * 256 blocks/image, 256 threads (8 wave32) each
    hipLaunchKernelGGL(st_cpab_kernel, dim3(2048), dim3(256), 0, stream, x, theta, out);
}